// GetCurvature_51805895524609
// MI455X (gfx1250) — compile-verified
//
#include <hip/hip_runtime.h>
#include <hip/hip_bf16.h>

typedef __attribute__((ext_vector_type(2))) float v2f;
typedef __attribute__((ext_vector_type(8))) float v8f;

#define IMG 1024
#define OH  128
#define NPATCH (OH * OH)
#define WAVES_PER_BLOCK 8

// Deterministic stand-in for the reference's tiny uniform regularizer noise.
__device__ __forceinline__ float hnoise(unsigned s) {
    s ^= s >> 16; s *= 2654435761u;
    s ^= s >> 13; s *= 1274126177u;
    s ^= s >> 16;
    return (float)(s & 0xFFFFFFu) * (1.0f / 16777216.0f);
}

// 4x4 inverse, Gauss-Jordan without pivoting (inputs are noise-regularized,
// matching the reference's inv(S22 + n)).
__device__ __forceinline__ void inv4(const float A[4][4], float R[4][4]) {
    float m[4][8];
#pragma unroll
    for (int i = 0; i < 4; i++) {
#pragma unroll
        for (int j = 0; j < 4; j++) { m[i][j] = A[i][j]; m[i][4 + j] = (i == j) ? 1.0f : 0.0f; }
    }
#pragma unroll
    for (int col = 0; col < 4; col++) {
        float ip = 1.0f / m[col][col];
#pragma unroll
        for (int j = 0; j < 8; j++) m[col][j] *= ip;
#pragma unroll
        for (int i = 0; i < 4; i++) {
            if (i == col) continue;
            float f = m[i][col];
#pragma unroll
            for (int j = 0; j < 8; j++) m[i][j] -= f * m[col][j];
        }
    }
#pragma unroll
    for (int i = 0; i < 4; i++)
#pragma unroll
        for (int j = 0; j < 4; j++) R[i][j] = m[i][4 + j];
}

__global__ __launch_bounds__(256) void curvature_kernel(const float* __restrict__ depth,
                                                        float* __restrict__ out) {
    __shared__ float lds[WAVES_PER_BLOCK * 64 * 16];   // 32 KB: D (64x16) per wave, reused for S

    const int lane  = threadIdx.x & 31;
    const int wave  = threadIdx.x >> 5;
    const int patch = blockIdx.x * WAVES_PER_BLOCK + wave;
    float* Dw = &lds[wave * 64 * 16];

    const int py = patch >> 7;
    const int px = patch & 127;
    const int yc = py * 8 + 3;
    const int xc = px * 8 + 3;
    const float dC = depth[yc * IMG + xc];

    // ---- build D rows (2 points per lane), 10 quadric features zero-padded to 16 ----
#pragma unroll
    for (int t = 0; t < 2; t++) {
        const int g  = lane * 2 + t;
        const int gy = g >> 3, gx = g & 7;
        const int r = py * 8 + gy - 1;          // padded image coordinate -> real coordinate
        const int c = px * 8 + gx - 1;
        const bool ok = (r >= 0) && (c >= 0);   // zero padding on top/left edge
        const float d = ok ? depth[r * IMG + c] : 0.0f;
        const float x = ok ? (float)c : 0.0f;
        const float y = ok ? (float)r : 0.0f;
        const float p1 = x - (float)xc;
        const float p2 = y - (float)yc;
        const float p3 = d - dC;
        float* row = &Dw[g * 16];
        row[0] = p1 * p1;    row[1] = p2 * p2;    row[2] = p3 * p3;
        row[3] = 2.f * p1 * p2; row[4] = 2.f * p1 * p3; row[5] = 2.f * p2 * p3;
        row[6] = 2.f * p1;   row[7] = 2.f * p2;   row[8] = 2.f * p3;
        row[9] = 0.f; row[10] = 0.f; row[11] = 0.f;
        row[12] = 0.f; row[13] = 0.f; row[14] = 0.f; row[15] = 0.f;
    }
    __syncthreads();

    // ---- S = D^T D via 16x V_WMMA_F32_16X16X4_F32 (K=64 in steps of 4) ----
    // A fragment (16x4 f32): VGPR0 lanes0-15 K=k0, lanes16-31 K=k0+2; VGPR1 K=k0+1 / K=k0+3.
    // With A = D^T, A[m,K] = D[K,m] = B[K,m]; the B 4x16 fragment mirrors A, so one
    // fragment feeds both operands and C accumulates the symmetric Gram matrix.
    const int half = lane >> 4;
    const int m    = lane & 15;
    v8f acc = {};
#pragma unroll
    for (int k = 0; k < 16; k++) {
        const int k0 = k * 4 + half * 2;
        v2f a;
        a.x = Dw[(k0 + 0) * 16 + m];
        a.y = Dw[(k0 + 1) * 16 + m];
        acc = __builtin_amdgcn_wmma_f32_16x16x4_f32(
            /*neg_a=*/false, a, /*neg_b=*/false, a,
            /*c_mod=*/(short)0, acc, /*reuse_a=*/false, /*reuse_b=*/false);
    }
    __syncthreads();

    // spill S (16x16, lanes0-15: M=r, lanes16-31: M=r+8; N = lane%16) into LDS
    float* Sw = Dw;
#pragma unroll
    for (int r8 = 0; r8 < 8; r8++)
        Sw[(r8 + half * 8) * 16 + m] = acc[r8];
    __syncthreads();

    // ---- scalar tail: one lane per patch ----
    if (lane == 0) {
        const unsigned pb = (unsigned)patch * 40503u;

        float S12[6][4];
#pragma unroll
        for (int i = 0; i < 6; i++)
#pragma unroll
            for (int j = 0; j < 4; j++) S12[i][j] = Sw[i * 16 + 6 + j];

        float S22a[4][4], S22b[4][4];
#pragma unroll
        for (int i = 0; i < 4; i++)
#pragma unroll
            for (int j = 0; j < 4; j++) {
                const float b = Sw[(6 + i) * 16 + 6 + j];
                S22a[i][j] = b + 1e-5f * hnoise(pb + (unsigned)(i * 4 + j) + 0x1000u);
                S22b[i][j] = b + 1e-5f * hnoise(pb + (unsigned)(i * 4 + j) + 0x2000u);
            }
        float I22a[4][4], I22b[4][4];
        inv4(S22a, I22a);
        inv4(S22b, I22b);

        // A6 = S11 - S12 * I22a * S12^T
        float T64[6][4];
#pragma unroll
        for (int i = 0; i < 6; i++)
#pragma unroll
            for (int l = 0; l < 4; l++) {
                float s = 0.f;
#pragma unroll
                for (int j = 0; j < 4; j++) s += S12[i][j] * I22a[j][l];
                T64[i][l] = s;
            }
        float A6[6][6];
#pragma unroll
        for (int i = 0; i < 6; i++)
#pragma unroll
            for (int mm = 0; mm < 6; mm++) {
                float s = 0.f;
#pragma unroll
                for (int l = 0; l < 4; l++) s += T64[i][l] * S12[mm][l];
                A6[i][mm] = Sw[i * 16 + mm] - s;
            }

        // M = inv(C) * A6 + n2;  inv(C) = blockdiag(0.5*(J-I), -0.25*I)  (closed form)
        float M6[6][6];
#pragma unroll
        for (int j = 0; j < 6; j++) {
            M6[0][j] = 0.5f * (A6[1][j] + A6[2][j]);
            M6[1][j] = 0.5f * (A6[0][j] + A6[2][j]);
            M6[2][j] = 0.5f * (A6[0][j] + A6[1][j]);
            M6[3][j] = -0.25f * A6[3][j];
            M6[4][j] = -0.25f * A6[4][j];
            M6[5][j] = -0.25f * A6[5][j];
        }
#pragma unroll
        for (int i = 0; i < 6; i++)
#pragma unroll
            for (int j = 0; j < 6; j++)
                M6[i][j] += 1e-4f * hnoise(pb + (unsigned)(i * 6 + j) + 0x3000u);

        // dominant-(real)-eigenvector of the 6x6 via shifted power iteration
        float shift = 0.f;
#pragma unroll
        for (int i = 0; i < 6; i++) {
            float rs = 0.f;
#pragma unroll
            for (int j = 0; j < 6; j++) rs += fabsf(M6[i][j]);
            shift = fmaxf(shift, rs);
        }
        float v[6] = {1.f, 1.f, 1.f, 1.f, 1.f, 1.f};
        for (int it = 0; it < 48; it++) {
            float w[6];
#pragma unroll
            for (int i = 0; i < 6; i++) {
                float s = shift * v[i];
#pragma unroll
                for (int j = 0; j < 6; j++) s += M6[i][j] * v[j];
                w[i] = s;
            }
            float nn = 1e-30f;
#pragma unroll
            for (int i = 0; i < 6; i++) nn += w[i] * w[i];
            const float inv = rsqrtf(nn);
#pragma unroll
            for (int i = 0; i < 6; i++) v[i] = w[i] * inv;
        }

        // vecs2 = -inv(S22+n3) * S12^T * vecs1
        float u[4];
#pragma unroll
        for (int j = 0; j < 4; j++) {
            float s = 0.f;
#pragma unroll
            for (int mm = 0; mm < 6; mm++) s += S12[mm][j] * v[mm];
            u[j] = s;
        }
        float q[10];
#pragma unroll
        for (int i = 0; i < 6; i++) q[i] = v[i];
#pragma unroll
        for (int i = 0; i < 4; i++) {
            float s = 0.f;
#pragma unroll
            for (int j = 0; j < 4; j++) s += I22b[i][j] * u[j];
            q[6 + i] = -s;
        }

        // quadric coefficient matrix (+ n4)
        float Co[4][4] = {
            {q[0], q[3], q[4], q[6]},
            {q[3], q[1], q[5], q[7]},
            {q[4], q[5], q[2], q[8]},
            {q[6], q[7], q[8], q[9]}};
#pragma unroll
        for (int i = 0; i < 4; i++)
#pragma unroll
            for (int j = 0; j < 4; j++)
                Co[i][j] += 1e-7f * hnoise(pb + (unsigned)(i * 4 + j) + 0x4000u);

        // centre = solve(-Co[:3,:3], q[6:9])  (Cramer)
        const float n00 = -Co[0][0], n01 = -Co[0][1], n02 = -Co[0][2];
        const float n10 = -Co[1][0], n11 = -Co[1][1], n12 = -Co[1][2];
        const float n20 = -Co[2][0], n21 = -Co[2][1], n22 = -Co[2][2];
        const float b0 = q[6], b1 = q[7], b2 = q[8];
        const float det = n00 * (n11 * n22 - n12 * n21)
                        - n01 * (n10 * n22 - n12 * n20)
                        + n02 * (n10 * n21 - n11 * n20);
        const float id = 1.0f / det;
        const float cx = id * (b0 * (n11 * n22 - n12 * n21)
                             - n01 * (b1 * n22 - n12 * b2)
                             + n02 * (b1 * n21 - n11 * b2));
        const float cy = id * (n00 * (b1 * n22 - n12 * b2)
                             - b0 * (n10 * n22 - n12 * n20)
                             + n02 * (n10 * b2 - b1 * n20));
        const float cz = id * (n00 * (n11 * b2 - b1 * n21)
                             - n01 * (n10 * b2 - b1 * n20)
                             + b0 * (n10 * n21 - n11 * n20));

        // R = T Co T^T with T = [[I3|0],[c,1]]; only R[3][3] is new vs Co[:3,:3]
        const float t4[4] = {cx, cy, cz, 1.0f};
        float R33 = 0.f;
#pragma unroll
        for (int j = 0; j < 4; j++) {
            float s = 0.f;
#pragma unroll
            for (int l = 0; l < 4; l++) s += Co[j][l] * t4[l];
            R33 += t4[j] * s;
        }
        const float sc = -1.0f / R33;
        const float a00 = Co[0][0] * sc, a11c = Co[1][1] * sc, a22c = Co[2][2] * sc;
        const float a01 = 0.5f * (Co[0][1] + Co[1][0]) * sc;
        const float a02 = 0.5f * (Co[0][2] + Co[2][0]) * sc;
        const float a12 = 0.5f * (Co[1][2] + Co[2][1]) * sc;

        // analytic symmetric 3x3 eigenvalues
        float e0, e1, e2;
        const float off = a01 * a01 + a02 * a02 + a12 * a12;
        if (off < 1e-30f) {
            e0 = a00; e1 = a11c; e2 = a22c;
        } else {
            const float qq = (a00 + a11c + a22c) * (1.0f / 3.0f);
            const float p2 = (a00 - qq) * (a00 - qq) + (a11c - qq) * (a11c - qq) +
                             (a22c - qq) * (a22c - qq) + 2.0f * off;
            const float pp = sqrtf(p2 * (1.0f / 6.0f));
            const float ipp = 1.0f / pp;
            const float c00 = (a00 - qq) * ipp, c11 = (a11c - qq) * ipp, c22 = (a22c - qq) * ipp;
            const float c01 = a01 * ipp, c02 = a02 * ipp, c12 = a12 * ipp;
            float r = 0.5f * (c00 * (c11 * c22 - c12 * c12)
                            - c01 * (c01 * c22 - c12 * c02)
                            + c02 * (c01 * c12 - c11 * c02));
            r = fminf(1.0f, fmaxf(-1.0f, r));
            const float phi = acosf(r) * (1.0f / 3.0f);
            e0 = qq + 2.0f * pp * cosf(phi);
            e2 = qq + 2.0f * pp * cosf(phi + 2.0943951023931953f);
            e1 = 3.0f * qq - e0 - e2;
        }

        const float ae0 = fmaxf(fabsf(e0), 1e-9f);
        const float ae1 = fmaxf(fabsf(e1), 1e-9f);
        const float ae2 = fmaxf(fabsf(e2), 1e-9f);
        const float amin = fminf(ae0, fminf(ae1, ae2));
        const float amax = fmaxf(ae0, fmaxf(ae1, ae2));
        // radii_i = sqrt(1/ae_i); simi = r_min/r_max = sqrt(amin/amax)
        out[patch] = sqrtf(amin / amax);
    }
}

extern "C" void kernel_launch(void* const* d_in, const int* in_sizes, int n_in,
                              void* d_out, int out_size, void* d_ws, size_t ws_size,
                              hipStream_t stream) {
    const float* depth = (const float*)d_in[0];
    float* out = (float*)d_out;
    (void)in_sizes; (void)n_in; (void)out_size; (void)d_ws; (void)ws_size;
    hipLaunchKernelGGL(curvature_kernel,
                       dim3(NPATCH / WAVES_PER_BLOCK), dim3(256), 0, stream,
                       depth, out);
}